// AbsolutePosEmb_42073499631774
// MI455X (gfx1250) — compile-verified
//
#include <hip/hip_runtime.h>

typedef __attribute__((ext_vector_type(2))) float v2f;
typedef __attribute__((ext_vector_type(8))) float v8f;

// Problem constants (B=4, N=8, H=32, W=32, D=64)
#define HW 1024   // H*W flattened positions
#define DD 64     // feature dim

// Each wave computes a 64x64 tile of logits[bn] = Q[bn] * (K[bn] + Emb)^T
// using 4x4 accumulators of V_WMMA_F32_16X16X4_F32.
__global__ __launch_bounds__(256)
void AbsolutePosEmb_wmma_kernel(const float* __restrict__ q,
                                const float* __restrict__ kk_,
                                const float* __restrict__ ph,
                                const float* __restrict__ pw,
                                float* __restrict__ out)
{
    const int wave = threadIdx.x >> 5;
    const int lane = threadIdx.x & 31;
    const int lr   = lane & 15;   // row within 16x16 fragment (M for A, N for B)
    const int kh   = lane >> 4;   // K-half select for 16x16x4 f32 fragments

    const int tile   = blockIdx.x * 8 + wave;   // 0..8191
    const int bn     = tile >> 8;               // 0..31
    const int t      = tile & 255;
    const int m_tile = (t >> 4) << 6;           // 0,64,...,960
    const int n_tile = (t & 15) << 6;

    const float* qbase = q   + (size_t)bn * HW * DD;
    const float* kbase = kk_ + (size_t)bn * HW * DD;

    // Per-lane row base pointers for the 4 M-subtiles / 4 N-subtiles.
    const float* aptr[4];
    const float* bptr[4];
    const float* phptr[4];
    const float* pwptr[4];
#pragma unroll
    for (int i = 0; i < 4; ++i) {
        const int rowA = m_tile + i * 16 + lr;
        const int rowB = n_tile + i * 16 + lr;
        aptr[i]  = qbase + (size_t)rowA * DD;
        bptr[i]  = kbase + (size_t)rowB * DD;
        phptr[i] = ph + (size_t)(rowB >> 5) * DD;   // p index = keypos / W
        pwptr[i] = pw + (size_t)(rowB & 31) * DD;   // q index = keypos % W
    }

    v8f acc[4][4];
    const v8f zero = {};
#pragma unroll
    for (int i = 0; i < 4; ++i)
#pragma unroll
        for (int j = 0; j < 4; ++j)
            acc[i][j] = zero;

    // K loop: D=64 in 16 steps of 4. Each lane loads the consecutive f32 pair
    // (K = kk + 2*kh, +1) of its row -> matches 16x16x4 f32 A/B VGPR layout.
    for (int kk = 0; kk < DD; kk += 4) {
        const int d = kk + 2 * kh;
        v2f a[4], b[4];
#pragma unroll
        for (int i = 0; i < 4; ++i)
            a[i] = *(const v2f*)(aptr[i] + d);
#pragma unroll
        for (int j = 0; j < 4; ++j) {
            const v2f kv = *(const v2f*)(bptr[j]  + d);
            const v2f pv = *(const v2f*)(phptr[j] + d);
            const v2f wv = *(const v2f*)(pwptr[j] + d);
            b[j] = kv + pv + wv;   // fused B' = K + ph + pw (emb folded into GEMM)
        }
#pragma unroll
        for (int i = 0; i < 4; ++i)
#pragma unroll
            for (int j = 0; j < 4; ++j)
                acc[i][j] = __builtin_amdgcn_wmma_f32_16x16x4_f32(
                    /*neg_a=*/false, a[i], /*neg_b=*/false, b[j],
                    /*c_mod=*/(short)0, acc[i][j],
                    /*reuse_a=*/false, /*reuse_b=*/false);
    }

    // Store: 16x16 f32 C/D layout -> VGPR r, lane l holds (M = r + 8*(l>=16), N = l&15).
    // Output is streamed once (134 MB total): use non-temporal stores.
    float* obase = out + (size_t)bn * HW * HW;
#pragma unroll
    for (int i = 0; i < 4; ++i) {
#pragma unroll
        for (int r = 0; r < 8; ++r) {
            const int m = m_tile + i * 16 + r + 8 * kh;
            float* orow = obase + (size_t)m * HW;
#pragma unroll
            for (int j = 0; j < 4; ++j) {
                const int n = n_tile + j * 16 + lr;
                __builtin_nontemporal_store(acc[i][j][r], orow + n);
            }
        }
    }
}

extern "C" void kernel_launch(void* const* d_in, const int* in_sizes, int n_in,
                              void* d_out, int out_size, void* d_ws, size_t ws_size,
                              hipStream_t stream) {
    const float* q  = (const float*)d_in[0];
    const float* k  = (const float*)d_in[1];
    const float* ph = (const float*)d_in[2];
    const float* pw = (const float*)d_in[3];
    float* out = (float*)d_out;

    // 32 bn-slices * (16x16) 64x64 tiles = 8192 wave-tiles, 8 waves per block.
    dim3 grid(1024), block(256);
    hipLaunchKernelGGL(AbsolutePosEmb_wmma_kernel, grid, block, 0, stream,
                       q, k, ph, pw, out);
}